// AugmentedChamferDistance_43800076485249
// MI455X (gfx1250) — compile-verified
//
#include <hip/hip_runtime.h>

typedef float v2f __attribute__((ext_vector_type(2)));
typedef float v8f __attribute__((ext_vector_type(8)));

#define BB 4
#define NN 4096
#define MM 4096
#define WAVES_PER_WG 8
#define ROWS_PER_WG (WAVES_PER_WG * 16)   // 128

// ---------------------------------------------------------------------------
// Init: seed col-min (and row-min, for safety) arrays with +inf bit pattern.
// ---------------------------------------------------------------------------
__global__ __launch_bounds__(256) void chamfer_init(unsigned* __restrict__ colmin,
                                                    unsigned* __restrict__ rowmin,
                                                    int n) {
  int i = blockIdx.x * blockDim.x + threadIdx.x;
  if (i < n) {
    colmin[i] = 0x7F800000u;   // +inf
    rowmin[i] = 0x7F800000u;
  }
}

// ---------------------------------------------------------------------------
// Tile kernel: each wave = 16 x-rows, sweeps all M y-points in 16-wide tiles.
// xy inner products via V_WMMA_F32_16X16X4_F32 (K=4: 3 coords + zero pad).
// Inner loop is software-pipelined (next y tile preloaded) and branch-free:
// both wave halves atomically ds_min into the same LDS column slot.
// ---------------------------------------------------------------------------
__global__ __launch_bounds__(256) void chamfer_tiles(const float* __restrict__ x,
                                                     const float* __restrict__ y,
                                                     float* __restrict__ rowmin,
                                                     unsigned* __restrict__ colmin_g) {
  __shared__ unsigned colmin_s[MM];   // 16 KB, shared by 8 waves

  const int tid  = threadIdx.x;
  const int lane = tid & 31;
  const int wave = tid >> 5;
  const int strips_per_batch = NN / ROWS_PER_WG;      // 32
  const int b        = blockIdx.x / strips_per_batch;
  const int strip    = blockIdx.x % strips_per_batch;
  const int row_base = strip * ROWS_PER_WG + wave * 16;

  // init LDS col-min to +inf
  for (int i = tid; i < MM; i += blockDim.x) colmin_s[i] = 0x7F800000u;
  __syncthreads();

  const int h = lane >> 4;     // half of wave (K-split per WMMA A/B layout)
  const int m = lane & 15;     // point index within 16-wide tile

  const float* xb = x + (size_t)b * NN * 3;
  const float* yb = y + (size_t)b * MM * 3;

  // A operand (16x4 f32): lane holds x-point row_base+m; v0=K(2h), v1=K(2h+1)
  const float px0 = xb[(row_base + m) * 3 + 0];
  const float px1 = xb[(row_base + m) * 3 + 1];
  const float px2 = xb[(row_base + m) * 3 + 2];
  const float x2  = px0 * px0 + px1 * px1 + px2 * px2;
  v2f a;
  a.x = h ? px2 : px0;   // K=0 / K=2
  a.y = h ? 0.0f : px1;  // K=1 / K=3 (pad)

  // |x|^2 for the 8 D-matrix rows this lane holds (rows k + 8h)
  float x2row[8];
#pragma unroll
  for (int k = 0; k < 8; ++k) x2row[k] = __shfl(x2, h * 8 + k, 32);

  float rmin[8];
#pragma unroll
  for (int k = 0; k < 8; ++k) rmin[k] = __builtin_inff();

  // software pipeline: preload first y tile
  float cy0 = yb[m * 3 + 0];
  float cy1 = yb[m * 3 + 1];
  float cy2 = yb[m * 3 + 2];

  for (int j0 = 0; j0 < MM; j0 += 16) {
    // issue next tile's load early (wraps to 0 on last iter; harmless)
    const int jn = (j0 + 16) & (MM - 1);
    const float ny0 = yb[(jn + m) * 3 + 0];
    const float ny1 = yb[(jn + m) * 3 + 1];
    const float ny2 = yb[(jn + m) * 3 + 2];

    // B operand (4x16 f32): column j0+m; v0=K(2h) row, v1=K(2h+1) row
    const float y2 = cy0 * cy0 + cy1 * cy1 + cy2 * cy2;
    v2f bop;
    bop.x = h ? cy2 : cy0;
    bop.y = h ? 0.0f : cy1;

    v8f c = {};
    c = __builtin_amdgcn_wmma_f32_16x16x4_f32(
        /*neg_a=*/false, a, /*neg_b=*/false, bop,
        /*c_mod=*/(short)0, c, /*reuse_a=*/false, /*reuse_b=*/false);

    // epilogue: d2 = max(x2 + y2 - 2*xy, 0); track row/col mins
    float cmin = __builtin_inff();
#pragma unroll
    for (int k = 0; k < 8; ++k) {
      float d2 = fmaxf(x2row[k] + y2 - 2.0f * c[k], 0.0f);
      rmin[k]  = fminf(rmin[k], d2);
      cmin     = fminf(cmin, d2);
    }
    // both halves min into the same slot: LDS atomic combines the 16 rows
    atomicMin(&colmin_s[j0 + m], __float_as_uint(cmin));

    cy0 = ny0; cy1 = ny1; cy2 = ny2;
  }

  // row-min: reduce across the 16 lanes of each half (rows k+8h)
#pragma unroll
  for (int mask = 1; mask < 16; mask <<= 1) {
#pragma unroll
    for (int k = 0; k < 8; ++k)
      rmin[k] = fminf(rmin[k], __shfl_xor(rmin[k], mask, 32));
  }
  if (m == 0) {
#pragma unroll
    for (int k = 0; k < 8; ++k)
      rowmin[(size_t)b * NN + row_base + h * 8 + k] = rmin[k];
  }

  __syncthreads();
  // flush block-local column mins to global (uint-min == float-min for >=0)
  for (int i = tid; i < MM; i += blockDim.x)
    atomicMin(&colmin_g[(size_t)b * MM + i], colmin_s[i]);
}

// ---------------------------------------------------------------------------
// Final reduction: means of both min arrays, output max of the two.
// ---------------------------------------------------------------------------
__global__ __launch_bounds__(512) void chamfer_reduce(const float* __restrict__ rowmin,
                                                      const float* __restrict__ colmin,
                                                      float* __restrict__ out) {
  __shared__ float sx[512];
  __shared__ float sy[512];
  float ax = 0.0f, ay = 0.0f;
  for (int i = threadIdx.x; i < BB * NN; i += blockDim.x) ax += rowmin[i];
  for (int i = threadIdx.x; i < BB * MM; i += blockDim.x) ay += colmin[i];
  sx[threadIdx.x] = ax;
  sy[threadIdx.x] = ay;
  __syncthreads();
  for (int s = blockDim.x >> 1; s > 0; s >>= 1) {
    if ((int)threadIdx.x < s) {
      sx[threadIdx.x] += sx[threadIdx.x + s];
      sy[threadIdx.x] += sy[threadIdx.x + s];
    }
    __syncthreads();
  }
  if (threadIdx.x == 0) {
    float x_to_y = sx[0] / (float)(BB * NN);
    float y_to_x = sy[0] / (float)(BB * MM);
    out[0] = fmaxf(x_to_y, y_to_x);
  }
}

// ---------------------------------------------------------------------------
extern "C" void kernel_launch(void* const* d_in, const int* in_sizes, int n_in,
                              void* d_out, int out_size, void* d_ws, size_t ws_size,
                              hipStream_t stream) {
  (void)in_sizes; (void)n_in; (void)out_size; (void)ws_size;
  const float* x = (const float*)d_in[0];   // [B, N, 3] fp32
  const float* y = (const float*)d_in[1];   // [B, M, 3] fp32

  float*    rowmin = (float*)d_ws;                                        // B*N floats
  unsigned* colmin = (unsigned*)((char*)d_ws + (size_t)BB * NN * sizeof(float)); // B*M

  chamfer_init<<<(BB * MM + 255) / 256, 256, 0, stream>>>(colmin, (unsigned*)rowmin,
                                                          BB * MM);
  chamfer_tiles<<<BB * (NN / ROWS_PER_WG), 256, 0, stream>>>(x, y, rowmin, colmin);
  chamfer_reduce<<<1, 512, 0, stream>>>(rowmin, (const float*)colmin, (float*)d_out);
}